// GraphTransformerLayer_61718680043592
// MI455X (gfx1250) — compile-verified
//
#include <hip/hip_runtime.h>
#include <math.h>

// ---------------------------------------------------------------------------
// GraphTransformerLayer on MI455X (gfx1250), fp32 WMMA (16x16x4) throughout.
// Memory-bound workload (~450MB HBM @ 23.3TB/s ~ 19us; 40 GFLOP is noise),
// so fp32 matrix ops give reference accuracy at no cost. Edge E-GEMM is fused
// into the attention kernel to avoid an 819MB E round-trip. Scatter segment-
// sums use native no-return global_atomic_add_f32 (L2-resident accumulators).
// ---------------------------------------------------------------------------

typedef __attribute__((ext_vector_type(2))) float v2f;
typedef __attribute__((ext_vector_type(8))) float v8f;

#define NNODE 50000
#define NEDGE 800000
#define DIM 128

static __device__ __forceinline__ v8f wmma4(v2f a, v2f b, v8f c) {
  // D = A(16x4) * B(4x16) + C, fp32, wave32
  return __builtin_amdgcn_wmma_f32_16x16x4_f32(false, a, false, b, (short)0, c,
                                               false, false);
}

static __device__ __forceinline__ v8f vzero8() {
  v8f z = {0.f, 0.f, 0.f, 0.f, 0.f, 0.f, 0.f, 0.f};
  return z;
}

// Force the native no-return HW float atomic (GLOBAL_ATOMIC_ADD_F32).
static __device__ __forceinline__ void atomAddF32(float* p, float v) {
  unsafeAtomicAdd(p, v);
}

// One wave computes a 16(M) x 64(N) tile of A[M,K]@B[K,N] (row-major).
// A-frag: lane m=lane&15 holds row m, K-pair r..r+1 with r = (lane>=16)?2:0.
// B-frag: lane holds col n=lane&15, same K split. Acc: v8f x 4 n-tiles.
static __device__ __forceinline__ void gemm_16x64(
    const float* __restrict__ A, int lda, const float* __restrict__ B, int ldb,
    int rowBase, int colBase, int K, v8f acc[4]) {
  const int lane = threadIdx.x;
  const int m = lane & 15;
  const int r = (lane >> 4) << 1;
  const float* Ar = A + (size_t)(rowBase + m) * lda + r;
  const float* Bc = B + (size_t)r * ldb + colBase + m;
  for (int k = 0; k < K; k += 4) {
    v2f a;
    a.x = Ar[k];
    a.y = Ar[k + 1];
    const float* Bk0 = Bc + (size_t)k * ldb;
    const float* Bk1 = Bk0 + ldb;
#pragma unroll
    for (int n = 0; n < 4; ++n) {
      v2f b;
      b.x = Bk0[n * 16];
      b.y = Bk1[n * 16];
      acc[n] = wmma4(a, b, acc[n]);
    }
  }
}

// ---------------------- QKV projection: [Q|K|V] = h @ [Wq|Wk|Wv] -----------
__global__ __launch_bounds__(32) void kQKV(const float* __restrict__ h,
                                           const float* __restrict__ Wq,
                                           const float* __restrict__ Wk,
                                           const float* __restrict__ Wv,
                                           float* __restrict__ Q,
                                           float* __restrict__ Kb,
                                           float* __restrict__ Vb) {
  const int rowBase = blockIdx.x * 16;
  const int colg = blockIdx.y * 64;  // 0..383
  const int w = colg >> 7;           // which weight matrix
  const float* B = (w == 0) ? Wq : ((w == 1) ? Wk : Wv);
  float* O = (w == 0) ? Q : ((w == 1) ? Kb : Vb);
  const int colB = colg & 127;

  v8f acc[4] = {vzero8(), vzero8(), vzero8(), vzero8()};
  gemm_16x64(h, DIM, B, DIM, rowBase, colB, DIM, acc);

  const int lane = threadIdx.x;
  const int rOff = (lane >> 4) * 8;
  const int bcol = lane & 15;
#pragma unroll
  for (int n = 0; n < 4; ++n)
#pragma unroll
    for (int i = 0; i < 8; ++i) {
      const int row = rowBase + i + rOff;
      O[(size_t)row * DIM + colB + n * 16 + bcol] = acc[n][i];
    }
}

// ------------- Edge kernel: fused E=e@We, score, exp, scatter --------------
// One wave per 16 edges. E tile via 256 f32 WMMAs; staged in LDS; then
// per-(edge,head) dots; then per-channel alpha & atomic scatter into Z/Hacc.
__global__ __launch_bounds__(32) void kEdge(
    const float* __restrict__ e, const int* __restrict__ src,
    const int* __restrict__ dst, const float* __restrict__ We,
    const float* __restrict__ Q, const float* __restrict__ Km,
    const float* __restrict__ V, float* __restrict__ Zacc,
    float* __restrict__ Hacc, int ne) {
  __shared__ float lds_E[16][132];  // row stride 132 floats (16B-aligned rows)
  __shared__ float lds_dot[16][4];

  const int base = blockIdx.x * 16;
  if (base >= ne) return;  // uniform
  const int lane = threadIdx.x;
  const int m = lane & 15;
  const int r = (lane >> 4) << 1;

  // ---- Phase 1: E(16x128) = e[base..base+15] @ We  (fp32 WMMA) ----
  v8f acc[8];
#pragma unroll
  for (int n = 0; n < 8; ++n) acc[n] = vzero8();
  {
    const float* Ar = e + (size_t)(base + m) * DIM + r;
    const float* Bc = We + (size_t)r * DIM + m;
    for (int k = 0; k < DIM; k += 4) {
      v2f a;
      a.x = Ar[k];
      a.y = Ar[k + 1];
      const float* Bk0 = Bc + (size_t)k * DIM;
      const float* Bk1 = Bk0 + DIM;
#pragma unroll
      for (int n = 0; n < 8; ++n) {
        v2f b;
        b.x = Bk0[n * 16];
        b.y = Bk1[n * 16];
        acc[n] = wmma4(a, b, acc[n]);
      }
    }
  }
  {
    const int rOff = (lane >> 4) * 8;
#pragma unroll
    for (int n = 0; n < 8; ++n)
#pragma unroll
      for (int i = 0; i < 8; ++i) lds_E[i + rOff][n * 16 + m] = acc[n][i];
  }
  __syncthreads();

  // ---- Phase 2: dot(m,h) = sum_k K[src[m]][32h+k] * Q[dst[m]][32h+k] ----
  // lane handles edge m=lane&15 and heads h = (lane>>4) and (lane>>4)+2.
  {
    const int em = base + m;
    const int sN = src[em];
    const int dN = dst[em];
    const float4* Kp = (const float4*)(Km + (size_t)sN * DIM);
    const float4* Qp = (const float4*)(Q + (size_t)dN * DIM);
#pragma unroll
    for (int hh = 0; hh < 2; ++hh) {
      const int hd = (lane >> 4) + hh * 2;
      float dot = 0.f;
#pragma unroll
      for (int j = 0; j < 8; ++j) {
        const float4 kv = Kp[hd * 8 + j];
        const float4 qv = Qp[hd * 8 + j];
        dot += kv.x * qv.x + kv.y * qv.y + kv.z * qv.z + kv.w * qv.w;
      }
      lds_dot[m][hd] = dot * 0.17677669529663687f;  // 1/sqrt(32)
    }
  }
  __syncthreads();

  // ---- Phase 3: alpha = exp(score*(1+E)); scatter alpha and V*alpha ----
  // lane owns cols 4*lane..4*lane+3 (single head: head = lane>>3).
  const int c0 = lane * 4;
  const int head = lane >> 3;
  for (int mm = 0; mm < 16; ++mm) {
    const int em = base + mm;
    const int sN = src[em];
    const int dN = dst[em];
    const float sd = lds_dot[mm][head];
    const float4 E4 = *(const float4*)&lds_E[mm][c0];
    const float4 V4 = *(const float4*)(V + (size_t)sN * DIM + c0);
    const float a0 = expf(sd * (1.f + E4.x));
    const float a1 = expf(sd * (1.f + E4.y));
    const float a2 = expf(sd * (1.f + E4.z));
    const float a3 = expf(sd * (1.f + E4.w));
    float* Zp = Zacc + (size_t)dN * DIM + c0;
    float* Hp = Hacc + (size_t)dN * DIM + c0;
    atomAddF32(Zp + 0, a0);
    atomAddF32(Zp + 1, a1);
    atomAddF32(Zp + 2, a2);
    atomAddF32(Zp + 3, a3);
    atomAddF32(Hp + 0, V4.x * a0);
    atomAddF32(Hp + 1, V4.y * a1);
    atomAddF32(Hp + 2, V4.z * a2);
    atomAddF32(Hp + 3, V4.w * a3);
  }
}

// ---- Attention out-proj: tmp = h + (Hacc/clip(Z)) @ Wlin + blin -----------
__global__ __launch_bounds__(32) void kProj(
    const float* __restrict__ Hacc, const float* __restrict__ Zacc,
    const float* __restrict__ Wlin, const float* __restrict__ blin,
    const float* __restrict__ h, float* __restrict__ out) {
  const int rowBase = blockIdx.x * 16;
  const int colB = blockIdx.y * 64;
  const int lane = threadIdx.x;
  const int m = lane & 15;
  const int r = (lane >> 4) << 1;

  v8f acc[4] = {vzero8(), vzero8(), vzero8(), vzero8()};
  const float* Ar = Hacc + (size_t)(rowBase + m) * DIM + r;
  const float* Zr = Zacc + (size_t)(rowBase + m) * DIM + r;
  const float* Bc = Wlin + (size_t)r * DIM + colB + m;
  for (int k = 0; k < DIM; k += 4) {
    v2f a;
    a.x = Ar[k] / fmaxf(Zr[k], 1e-6f);
    a.y = Ar[k + 1] / fmaxf(Zr[k + 1], 1e-6f);
    const float* Bk0 = Bc + (size_t)k * DIM;
    const float* Bk1 = Bk0 + DIM;
#pragma unroll
    for (int n = 0; n < 4; ++n) {
      v2f b;
      b.x = Bk0[n * 16];
      b.y = Bk1[n * 16];
      acc[n] = wmma4(a, b, acc[n]);
    }
  }
  const int rOff = (lane >> 4) * 8;
  const int bcol = lane & 15;
#pragma unroll
  for (int n = 0; n < 4; ++n)
#pragma unroll
    for (int i = 0; i < 8; ++i) {
      const int row = rowBase + i + rOff;
      const int col = colB + n * 16 + bcol;
      out[(size_t)row * DIM + col] =
          acc[n][i] + h[(size_t)row * DIM + col] + blin[col];
    }
}

// -------------------- FFN1: f = relu(h1 @ Wf1 + bf1), N=256 ----------------
__global__ __launch_bounds__(32) void kFFN1(const float* __restrict__ h1,
                                            const float* __restrict__ Wf1,
                                            const float* __restrict__ bf1,
                                            float* __restrict__ f) {
  const int rowBase = blockIdx.x * 16;
  const int colB = blockIdx.y * 64;
  v8f acc[4] = {vzero8(), vzero8(), vzero8(), vzero8()};
  gemm_16x64(h1, DIM, Wf1, 2 * DIM, rowBase, colB, DIM, acc);
  const int lane = threadIdx.x;
  const int rOff = (lane >> 4) * 8;
  const int bcol = lane & 15;
#pragma unroll
  for (int n = 0; n < 4; ++n)
#pragma unroll
    for (int i = 0; i < 8; ++i) {
      const int row = rowBase + i + rOff;
      const int col = colB + n * 16 + bcol;
      f[(size_t)row * (2 * DIM) + col] = fmaxf(acc[n][i] + bf1[col], 0.f);
    }
}

// ---------------- FFN2: tmp2 = h1 + f @ Wf2 + bf2, K=256 -------------------
__global__ __launch_bounds__(32) void kFFN2(const float* __restrict__ f,
                                            const float* __restrict__ Wf2,
                                            const float* __restrict__ bf2,
                                            const float* __restrict__ h1,
                                            float* __restrict__ out) {
  const int rowBase = blockIdx.x * 16;
  const int colB = blockIdx.y * 64;
  v8f acc[4] = {vzero8(), vzero8(), vzero8(), vzero8()};
  gemm_16x64(f, 2 * DIM, Wf2, DIM, rowBase, colB, 2 * DIM, acc);
  const int lane = threadIdx.x;
  const int rOff = (lane >> 4) * 8;
  const int bcol = lane & 15;
#pragma unroll
  for (int n = 0; n < 4; ++n)
#pragma unroll
    for (int i = 0; i < 8; ++i) {
      const int row = rowBase + i + rOff;
      const int col = colB + n * 16 + bcol;
      out[(size_t)row * DIM + col] =
          acc[n][i] + bf2[col] + h1[(size_t)row * DIM + col];
    }
}

// ------------------------- LayerNorm: one wave per row ---------------------
__global__ __launch_bounds__(256) void kLN(const float* __restrict__ in,
                                           const float* __restrict__ g,
                                           const float* __restrict__ b,
                                           float* __restrict__ out, int nrows) {
  const int row = blockIdx.x * 8 + (threadIdx.x >> 5);
  if (row >= nrows) return;
  const int lane = threadIdx.x & 31;
  const float4 x = *(const float4*)(in + (size_t)row * DIM + lane * 4);
  float s = x.x + x.y + x.z + x.w;
#pragma unroll
  for (int off = 16; off > 0; off >>= 1) s += __shfl_xor(s, off, 32);
  const float mu = s * (1.f / 128.f);
  const float d0 = x.x - mu, d1 = x.y - mu, d2 = x.z - mu, d3 = x.w - mu;
  float v = d0 * d0 + d1 * d1 + d2 * d2 + d3 * d3;
#pragma unroll
  for (int off = 16; off > 0; off >>= 1) v += __shfl_xor(v, off, 32);
  const float inv = rsqrtf(v * (1.f / 128.f) + 1e-5f);
  const float4 gg = *(const float4*)(g + lane * 4);
  const float4 bb = *(const float4*)(b + lane * 4);
  float4 o;
  o.x = d0 * inv * gg.x + bb.x;
  o.y = d1 * inv * gg.y + bb.y;
  o.z = d2 * inv * gg.z + bb.z;
  o.w = d3 * inv * gg.w + bb.w;
  *(float4*)(out + (size_t)row * DIM + lane * 4) = o;
}

// ---------------------------------------------------------------------------
extern "C" void kernel_launch(void* const* d_in, const int* in_sizes, int n_in,
                              void* d_out, int out_size, void* d_ws,
                              size_t ws_size, hipStream_t stream) {
  const float* h = (const float*)d_in[0];
  const float* e = (const float*)d_in[1];
  const int* src = (const int*)d_in[2];
  const int* dst = (const int*)d_in[3];
  const float* Wq = (const float*)d_in[4];
  const float* Wk = (const float*)d_in[5];
  const float* Wv = (const float*)d_in[6];
  const float* We = (const float*)d_in[7];
  const float* Wlin = (const float*)d_in[8];
  const float* blin = (const float*)d_in[9];
  const float* ln1g = (const float*)d_in[10];
  const float* ln1b = (const float*)d_in[11];
  const float* Wf1 = (const float*)d_in[12];
  const float* bf1 = (const float*)d_in[13];
  const float* Wf2 = (const float*)d_in[14];
  const float* bf2 = (const float*)d_in[15];
  const float* ln2g = (const float*)d_in[16];
  const float* ln2b = (const float*)d_in[17];

  // ws layout: 5 regions of N*D floats (128MB total), with lifetime reuse.
  float* ws = (float*)d_ws;
  const size_t NB = (size_t)NNODE * DIM;  // 6.4M floats
  float* Q = ws + 0 * NB;                 // later: tmp (pre-LN1)
  float* Kb = ws + 1 * NB;                // later: tmp2 (pre-LN2)
  float* Vb = ws + 2 * NB;                // later: h1
  float* Zacc = ws + 3 * NB;              // later: f[:25000, :]
  float* Hacc = ws + 4 * NB;              // later: f[25000:, :]
  float* tmp = Q;
  float* tmp2 = Kb;
  float* h1 = Vb;
  float* fbuf = Zacc;  // N x 256, contiguous across regions 3+4

  const int MT = NNODE / 16;  // 3125 row tiles (exact)

  hipMemsetAsync(Zacc, 0, 2 * NB * sizeof(float), stream);
  kQKV<<<dim3(MT, 6), 32, 0, stream>>>(h, Wq, Wk, Wv, Q, Kb, Vb);
  kEdge<<<NEDGE / 16, 32, 0, stream>>>(e, src, dst, We, Q, Kb, Vb, Zacc, Hacc,
                                       NEDGE);
  kProj<<<dim3(MT, 2), 32, 0, stream>>>(Hacc, Zacc, Wlin, blin, h, tmp);
  kLN<<<NNODE / 8, 256, 0, stream>>>(tmp, ln1g, ln1b, h1, NNODE);
  kFFN1<<<dim3(MT, 4), 32, 0, stream>>>(h1, Wf1, bf1, fbuf);
  kFFN2<<<dim3(MT, 2), 32, 0, stream>>>(fbuf, Wf2, bf2, h1, tmp2);
  kLN<<<NNODE / 8, 256, 0, stream>>>(tmp2, ln2g, ln2b, (float*)d_out, NNODE);
}